// GraphQNN__gen246_65481071396014
// MI455X (gfx1250) — compile-verified
//
#include <hip/hip_runtime.h>
#include <hip/hip_bf16.h>

typedef float v2f __attribute__((ext_vector_type(2)));
typedef float v8f __attribute__((ext_vector_type(8)));

#define NQ 22
#define DIMN (1u << NQ)
#define NTILES (1u << 14)          // 2^22 / 256 amplitudes per 16x16 tile
#define GATE_GRID 1024             // blocks (8 waves each -> 8192 waves, 2 tiles/wave)
#define RED_BLOCKS 1024

static __device__ __forceinline__ v8f wmma4(v2f a, v2f b, v8f c) {
    // D = A(16x4,f32) * B(4x16,f32) + C(16x16,f32)
    return __builtin_amdgcn_wmma_f32_16x16x4_f32(false, a, false, b, (short)0, c,
                                                 false, false);
}

// ---------------------------------------------------------------------------
// Gate-matrix prep: 4 layers x 6 groups, each a 16x16 complex unitary.
// Group qubit lists (wire q0 = largest stride = MSB of the 4-bit group index).
// Group 5 = qubits {0,1} padded with identity on the low two group bits.
// U_q = RZ(tz) * RY(ty) * RX(tx) fused 2x2; group = Kronecker of four 2x2s.
// ---------------------------------------------------------------------------
__global__ void qprep(const float* __restrict__ params, float* __restrict__ gates) {
    const int blk = blockIdx.x;            // 0..23
    const int l = blk / 6, g = blk % 6;
    const int e = threadIdx.x;             // 0..255
    const int a = e >> 4, b = e & 15;
    const int groups[6][4] = {{2,3,4,5},{6,7,8,9},{10,11,12,13},
                              {14,15,16,17},{18,19,20,21},{0,1,-1,-1}};
    float ure = 1.0f, uim = 0.0f;
    #pragma unroll
    for (int j = 0; j < 4; ++j) {
        int qq = groups[g][j];
        int aj = (a >> (3 - j)) & 1;
        int bj = (b >> (3 - j)) & 1;
        float er, ei;
        if (qq < 0) {                       // identity factor
            er = (aj == bj) ? 1.0f : 0.0f; ei = 0.0f;
        } else {
            float tx = params[(l*NQ + qq)*3 + 0];
            float ty = params[(l*NQ + qq)*3 + 1];
            float tz = params[(l*NQ + qq)*3 + 2];
            float cx = cosf(0.5f*tx), sx = sinf(0.5f*tx);
            float cy = cosf(0.5f*ty), sy = sinf(0.5f*ty);
            float cz = cosf(0.5f*tz), sz = sinf(0.5f*tz);
            // M = RY*RX
            float mr, mi;
            if      (aj == 0 && bj == 0) { mr =  cy*cx; mi =  sy*sx; }
            else if (aj == 0)            { mr = -sy*cx; mi = -cy*sx; }
            else if (bj == 0)            { mr =  sy*cx; mi = -cy*sx; }
            else                         { mr =  cy*cx; mi = -sy*sx; }
            // U = RZ*M : row phase e^{-i tz/2} (a=0) / e^{+i tz/2} (a=1)
            float pr = cz, pi = (aj == 0) ? -sz : sz;
            er = pr*mr - pi*mi;
            ei = pr*mi + pi*mr;
        }
        float nr = ure*er - uim*ei;
        float ni = ure*ei + uim*er;
        ure = nr; uim = ni;
    }
    gates[blk*512 + e]       = ure;   // re(16x16), row-major a*16+b
    gates[blk*512 + 256 + e] = uim;   // im(16x16)
}

// ---------------------------------------------------------------------------
// Strided 4-qubit group pass (group qubits q..q+3, inner size J=2^(18-q)>=16).
// Tile = 16 group-index rows (stride J) x 16 contiguous columns.
// One wave per tile; 16 x V_WMMA_F32_16X16X4_F32 per tile (complex 16x16 GEMM).
// If permute!=0, loads gather through the Gray map g(x)=x^(x>>1), which is the
// composed CNOT chain of the previous layer (folded for free).
// ---------------------------------------------------------------------------
__global__ void qgate_strided(const float* __restrict__ srcRe, const float* __restrict__ srcIm,
                              float* __restrict__ dstRe, float* __restrict__ dstIm,
                              const float* __restrict__ mat, int q, int permute) {
    const int lane = threadIdx.x & 31;
    const int half = lane >> 4;
    const int l16  = lane & 15;
    const unsigned J = 1u << (18 - q);
    const unsigned jbShift = (unsigned)(14 - q);
    const unsigned jbMask  = (1u << jbShift) - 1u;

    // A operands: uniform over tiles, held in registers across the tile loop
    v2f aRe[4], aIm[4], aImN[4];
    #pragma unroll
    for (int k = 0; k < 4; ++k) {
        int col = 4*k + 2*half;
        aRe[k].x = mat[l16*16 + col];
        aRe[k].y = mat[l16*16 + col + 1];
        aIm[k].x = mat[256 + l16*16 + col];
        aIm[k].y = mat[256 + l16*16 + col + 1];
        aImN[k]  = -aIm[k];
    }

    const unsigned waveId = blockIdx.x * (blockDim.x >> 5) + (threadIdx.x >> 5);
    const unsigned nWaves = gridDim.x * (blockDim.x >> 5);

    for (unsigned t = waveId; t < NTILES; t += nWaves) {
        unsigned i    = t >> jbShift;
        unsigned jb   = t & jbMask;
        unsigned base = i * (J << 4) + (jb << 4);

        v2f bRe[4], bIm[4];
        #pragma unroll
        for (int k = 0; k < 4; ++k) {
            unsigned r0 = 4u*k + 2u*half;
            unsigned a0 = base + r0*J + (unsigned)l16;
            unsigned a1 = a0 + J;
            if (permute) { a0 ^= (a0 >> 1); a1 ^= (a1 >> 1); }
            bRe[k].x = srcRe[a0]; bRe[k].y = srcRe[a1];
            bIm[k].x = srcIm[a0]; bIm[k].y = srcIm[a1];
        }

        v8f cRe = {}; v8f cIm = {};
        #pragma unroll
        for (int k = 0; k < 4; ++k) cRe = wmma4(aRe[k],  bRe[k], cRe);
        #pragma unroll
        for (int k = 0; k < 4; ++k) cRe = wmma4(aImN[k], bIm[k], cRe);
        #pragma unroll
        for (int k = 0; k < 4; ++k) cIm = wmma4(aRe[k],  bIm[k], cIm);
        #pragma unroll
        for (int k = 0; k < 4; ++k) cIm = wmma4(aIm[k],  bRe[k], cIm);

        #pragma unroll
        for (int r = 0; r < 8; ++r) {
            unsigned row = (unsigned)r + 8u*half;
            unsigned ad  = base + row*J + (unsigned)l16;
            dstRe[ad] = cRe[r];
            dstIm[ad] = cIm[r];
        }
    }
}

// ---------------------------------------------------------------------------
// Low-group pass: qubits {18..21} = the 4 LSBs. Tile = 256 contiguous amps,
// B[b,c] = s[base + c*16 + b] (transposed tile). Always in-place, no permute.
// ---------------------------------------------------------------------------
__global__ void qgate_low(float* __restrict__ stRe, float* __restrict__ stIm,
                          const float* __restrict__ mat) {
    const int lane = threadIdx.x & 31;
    const int half = lane >> 4;
    const int l16  = lane & 15;

    v2f aRe[4], aIm[4], aImN[4];
    #pragma unroll
    for (int k = 0; k < 4; ++k) {
        int col = 4*k + 2*half;
        aRe[k].x = mat[l16*16 + col];
        aRe[k].y = mat[l16*16 + col + 1];
        aIm[k].x = mat[256 + l16*16 + col];
        aIm[k].y = mat[256 + l16*16 + col + 1];
        aImN[k]  = -aIm[k];
    }

    const unsigned waveId = blockIdx.x * (blockDim.x >> 5) + (threadIdx.x >> 5);
    const unsigned nWaves = gridDim.x * (blockDim.x >> 5);

    for (unsigned t = waveId; t < NTILES; t += nWaves) {
        unsigned base = t << 8;                    // 256 contiguous amplitudes

        v2f bRe[4], bIm[4];
        #pragma unroll
        for (int k = 0; k < 4; ++k) {
            unsigned r0 = 4u*k + 2u*half;
            unsigned a0 = base + (unsigned)l16*16u + r0;   // B[b,c] = s[base+c*16+b]
            bRe[k].x = stRe[a0]; bRe[k].y = stRe[a0 + 1];
            bIm[k].x = stIm[a0]; bIm[k].y = stIm[a0 + 1];
        }

        v8f cRe = {}; v8f cIm = {};
        #pragma unroll
        for (int k = 0; k < 4; ++k) cRe = wmma4(aRe[k],  bRe[k], cRe);
        #pragma unroll
        for (int k = 0; k < 4; ++k) cRe = wmma4(aImN[k], bIm[k], cRe);
        #pragma unroll
        for (int k = 0; k < 4; ++k) cIm = wmma4(aRe[k],  bIm[k], cIm);
        #pragma unroll
        for (int k = 0; k < 4; ++k) cIm = wmma4(aIm[k],  bRe[k], cIm);

        #pragma unroll
        for (int r = 0; r < 8; ++r) {
            unsigned row = (unsigned)r + 8u*half;
            unsigned ad  = base + (unsigned)l16*16u + row;
            stRe[ad] = cRe[r];
            stIm[ad] = cIm[r];
        }
    }
}

// ---------------------------------------------------------------------------
// Measurement: <Z_q> on the post-CNOT-chain state equals signed prob sums on
// the pre-permutation state with sign_q = 1-2*parity(idx >> (21-q)).
// Also accumulates total norm (index 22) so normalization can be deferred.
// Deterministic two-stage reduction (no float atomics).
// ---------------------------------------------------------------------------
__global__ void qreduce(const float* __restrict__ re, const float* __restrict__ im,
                        float* __restrict__ partial) {
    float acc[23];
    #pragma unroll
    for (int a = 0; a < 23; ++a) acc[a] = 0.0f;
    unsigned tid = blockIdx.x * blockDim.x + threadIdx.x;
    unsigned stride = gridDim.x * blockDim.x;
    for (unsigned idx = tid; idx < DIMN; idx += stride) {
        float r = re[idx], i = im[idx];
        float p = r*r + i*i;
        acc[22] += p;
        unsigned P = 0;
        #pragma unroll
        for (int q = 0; q < 22; ++q) {
            P ^= (idx >> (21 - q)) & 1u;
            acc[q] += P ? -p : p;
        }
    }
    __shared__ float red[256];
    for (int a = 0; a < 23; ++a) {
        red[threadIdx.x] = acc[a];
        __syncthreads();
        for (int s = 128; s > 0; s >>= 1) {
            if ((int)threadIdx.x < s) red[threadIdx.x] += red[threadIdx.x + s];
            __syncthreads();
        }
        if (threadIdx.x == 0) partial[blockIdx.x*23 + a] = red[0];
        __syncthreads();
    }
}

__global__ void qfinal(const float* __restrict__ partial, float* __restrict__ out,
                       int nblocks) {
    int q = threadIdx.x;
    if (q >= 22) return;
    double s = 0.0, n = 0.0;
    for (int b = 0; b < nblocks; ++b) {
        s += (double)partial[b*23 + q];
        n += (double)partial[b*23 + 22];
    }
    out[q] = (float)(s / n);
}

// ---------------------------------------------------------------------------
extern "C" void kernel_launch(void* const* d_in, const int* in_sizes, int n_in,
                              void* d_out, int out_size, void* d_ws, size_t ws_size,
                              hipStream_t stream) {
    const float* params = (const float*)d_in[0];
    const float* inRe   = (const float*)d_in[1];
    const float* inIm   = (const float*)d_in[2];
    float* out = (float*)d_out;

    char* ws = (char*)d_ws;
    const size_t D4 = (size_t)DIMN * 4u;
    float* Are = (float*)(ws);
    float* Aim = (float*)(ws + D4);
    float* Bre = (float*)(ws + 2*D4);
    float* Bim = (float*)(ws + 3*D4);
    float* gates   = (float*)(ws + 4*D4);                  // 24 * 512 floats
    float* partial = (float*)(ws + 4*D4 + 24*512*4);       // RED_BLOCKS * 23 floats

    qprep<<<24, 256, 0, stream>>>(params, gates);

    const int qlist[4] = {2, 6, 10, 14};
    float* bufRe[2] = {Are, Bre};
    float* bufIm[2] = {Aim, Bim};
    const float* prevRe = inRe;
    const float* prevIm = inIm;

    for (int l = 0; l < 4; ++l) {
        float* curRe = bufRe[l & 1];
        float* curIm = bufIm[l & 1];
        // pass 0: group {2-5}, gathers previous layer's CNOT permutation (l>0)
        qgate_strided<<<GATE_GRID, 256, 0, stream>>>(prevRe, prevIm, curRe, curIm,
                                                     gates + (l*6 + 0)*512, 2,
                                                     (l > 0) ? 1 : 0);
        // passes 1..3: groups {6-9},{10-13},{14-17}, in place
        for (int gi = 1; gi <= 3; ++gi)
            qgate_strided<<<GATE_GRID, 256, 0, stream>>>(curRe, curIm, curRe, curIm,
                                                         gates + (l*6 + gi)*512,
                                                         qlist[gi], 0);
        // pass 4: group {18-21} (low bits), in place
        qgate_low<<<GATE_GRID, 256, 0, stream>>>(curRe, curIm, gates + (l*6 + 4)*512);
        // pass 5: qubits {0,1} padded as {0-3} with identity, in place
        qgate_strided<<<GATE_GRID, 256, 0, stream>>>(curRe, curIm, curRe, curIm,
                                                     gates + (l*6 + 5)*512, 0, 0);
        prevRe = curRe; prevIm = curIm;
    }

    // Final CNOT chain folded into signs; normalization deferred to division.
    qreduce<<<RED_BLOCKS, 256, 0, stream>>>(prevRe, prevIm, partial);
    qfinal<<<1, 32, 0, stream>>>(partial, out, RED_BLOCKS);
}